// GsplatRGB_20444044329830
// MI455X (gfx1250) — compile-verified
//
#include <hip/hip_runtime.h>
#include <hip/hip_bf16.h>

#define FX      1000.0f
#define FY      1000.0f
#define HALF_W  512.0f   // WIDTH/2
#define HALF_H  512.0f   // HEIGHT/2
#define N_GAUSS 512
#define B_POSE  4
#define TILE_H  128      // HU-HL
#define TILE_W  128      // WU-WL

// -0.5 * log2(e): folded into the coefficients so the WMMA accumulator is
// directly the exp2 argument (epilogue needs no scaling multiplies).
#define NEG_HALF_LOG2E (-0.72134752044448170368f)

typedef __attribute__((ext_vector_type(2))) float v2f;
typedef __attribute__((ext_vector_type(8))) float v8f;

// ---------------------------------------------------------------------------
// Kernel 1: per-(pose, gaussian) coefficient precompute.
// Collapses quat->rot, world->cam, Jacobian, det into 6 coefficients of the
// bilinear form  -0.5*log2e*prob = f(x,y) . g(b,n),  f = [x^2,y^2,xy,x,y,1].
// Output layout: G[b][n][8]; g6=g7=0 padding doubles as the zero source for
// the high-lane B-fragment loads in kernel 2.
// ---------------------------------------------------------------------------
__global__ __launch_bounds__(256) void gsplat_coeff_kernel(
    const float* __restrict__ pose,    // [B][4][4]
    const float* __restrict__ means,   // [N][3]
    const float* __restrict__ quats,   // [N][4]
    const float* __restrict__ scales,  // [N][3]
    float* __restrict__ G)             // [B][N][8]
{
    const int idx = blockIdx.x * blockDim.x + threadIdx.x;
    if (idx >= B_POSE * N_GAUSS) return;
    const int b = idx >> 9;          // / N_GAUSS
    const int n = idx & (N_GAUSS - 1);

    const float* P = pose + b * 16;
    const float mex = means[n * 3 + 0];
    const float mey = means[n * 3 + 1];
    const float mez = means[n * 3 + 2];

    float qw = quats[n * 4 + 0], qx = quats[n * 4 + 1];
    float qy = quats[n * 4 + 2], qz = quats[n * 4 + 3];
    const float qinv = rsqrtf(qw * qw + qx * qx + qy * qy + qz * qz);
    qw *= qinv; qx *= qinv; qy *= qinv; qz *= qinv;

    const float R00 = 1.0f - 2.0f * (qy * qy + qz * qz);
    const float R01 = 2.0f * (qx * qy - qw * qz);
    const float R02 = 2.0f * (qx * qz + qw * qy);
    const float R10 = 2.0f * (qx * qy + qw * qz);
    const float R11 = 1.0f - 2.0f * (qx * qx + qz * qz);
    const float R12 = 2.0f * (qy * qz - qw * qx);
    const float R20 = 2.0f * (qx * qz - qw * qy);
    const float R21 = 2.0f * (qy * qz + qw * qx);
    const float R22 = 1.0f - 2.0f * (qx * qx + qy * qy);

    const float s0 = scales[n * 3 + 0];
    const float s1 = scales[n * 3 + 1];
    const float s2 = scales[n * 3 + 2];

    // Ms_world = R with column scaling
    const float W00 = R00 * s0, W01 = R01 * s1, W02 = R02 * s2;
    const float W10 = R10 * s0, W11 = R11 * s1, W12 = R12 * s2;
    const float W20 = R20 * s0, W21 = R21 * s1, W22 = R22 * s2;

    const float P00 = P[0],  P01 = P[1],  P02 = P[2],  P03 = P[3];
    const float P10 = P[4],  P11 = P[5],  P12 = P[6],  P13 = P[7];
    const float P20 = P[8],  P21 = P[9],  P22 = P[10], P23 = P[11];

    // camera-space mean
    const float u = P00 * mex + P01 * mey + P02 * mez + P03;
    const float v = P10 * mex + P11 * mey + P12 * mez + P13;
    const float d = P20 * mex + P21 * mey + P22 * mez + P23;

    // Ms_cam = pose[:3,:3] @ Ms_world
    const float C00 = P00 * W00 + P01 * W10 + P02 * W20;
    const float C01 = P00 * W01 + P01 * W11 + P02 * W21;
    const float C02 = P00 * W02 + P01 * W12 + P02 * W22;
    const float C10 = P10 * W00 + P11 * W10 + P12 * W20;
    const float C11 = P10 * W01 + P11 * W11 + P12 * W21;
    const float C12 = P10 * W02 + P11 * W12 + P12 * W22;
    const float C20 = P20 * W00 + P21 * W10 + P22 * W20;
    const float C21 = P20 * W01 + P21 * W11 + P22 * W21;
    const float C22 = P20 * W02 + P21 * W12 + P22 * W22;

    // Mp = J @ Ms_cam, J = [[FX*d, 0, -FX*u], [0, FY*d, -FY*v]]
    const float t00 = FX * (d * C00 - u * C20);
    const float t01 = FX * (d * C01 - u * C21);
    const float t02 = FX * (d * C02 - u * C22);
    const float t10 = FY * (d * C10 - v * C20);
    const float t11 = FY * (d * C11 - v * C21);
    const float t12 = FY * (d * C12 - v * C22);

    const float e0 = t00 * t11 - t01 * t10;
    const float e1 = t00 * t12 - t02 * t10;
    const float e2 = t01 * t12 - t02 * t11;
    // fold -0.5*log2(e) into 1/det so the GEMM emits the exp2 argument
    const float rdet = NEG_HALF_LOG2E / (e0 * e0 + e1 * e1 + e2 * e2);

    const float Aq = (t10 * t10 + t11 * t11 + t12 * t12) * rdet;
    const float Bq = -2.0f * (t00 * t10 + t01 * t11 + t02 * t12) * rdet;
    const float Cq = (t00 * t00 + t01 * t01 + t02 * t02) * rdet;

    const float mx = FX * u + HALF_W * d;   // pixel-space mean (homog)
    const float my = FY * v + HALF_H * d;
    const float d2 = d * d, d3 = d2 * d, d4 = d2 * d2;

    float* g = G + ((size_t)idx << 3);
    g[0] = Aq * d4;                               // x^2
    g[1] = Cq * d4;                               // y^2
    g[2] = Bq * d4;                               // x*y
    g[3] = -(2.0f * Aq * mx + Bq * my) * d3;      // x
    g[4] = -(2.0f * Cq * my + Bq * mx) * d3;      // y
    g[5] = (Aq * mx * mx + Bq * mx * my + Cq * my * my) * d2;  // 1
    g[6] = 0.0f;                                  // K padding / hi-lane zeros
    g[7] = 0.0f;
}

// ---------------------------------------------------------------------------
// Kernel 2: main evaluation as a K=8 fp32 WMMA GEMM + exp2 epilogue.
// Block = 256 threads (8 waves). Each block owns one (pose, pixel-row-tile)
// of 16 consecutive x at fixed y; wave w handles gaussian tiles w*4..w*4+3.
// A-frag (16x4 f32): lanes 0-15 hold K0,K1; lanes 16-31 hold K2,K3.
// B-frag (4x16 f32): lanes 0-15 col n=lane K0,K1; lanes 16-31 K2,K3.
// D (16x16 f32): vgpr r = row r (lanes 0-15) / r+8 (lanes 16-31), col lane%16.
// All lane-half selection is branchless (cndmask) so the unrolled B-fragment
// loads issue as one clause; the accumulator is already the exp2 argument and
// the epilogue uses the raw v_exp_f32 (no libm range fixup).
// ---------------------------------------------------------------------------
__global__ __launch_bounds__(256) void gsplat_wmma_kernel(
    const float* __restrict__ G,     // [B][N][8]
    const float* __restrict__ opac,  // [N]
    float* __restrict__ out)         // [B][128][128][N]
{
    const int tile   = blockIdx.x;          // 0 .. B*1024-1
    const int b      = tile >> 10;
    const int pt     = tile & 1023;
    const int y      = pt >> 3;             // pixel row 0..127
    const int x0     = (pt & 7) << 4;       // 16-pixel x tile
    const int wave   = threadIdx.x >> 5;
    const int lane   = threadIdx.x & 31;
    const int lane16 = lane & 15;
    const bool hi    = lane >= 16;

    const float* gpose = G + ((size_t)b << 12);   // b * 512 * 8

    // pull this pose's 16KB coefficient table toward the caches
    __builtin_prefetch(((const char*)gpose) + (size_t)threadIdx.x * 64, 0, 1);

    // Pixel features for A-fragment row m = lane%16 (x = x0+m), f(x,y):
    //   K0..K3 = [x^2, y^2, x*y, x] ; K4..K7 = [y, 1, 0, 0]
    const float xf = (float)(x0 + lane16);
    const float yf = (float)y;
    v2f a0, a1;
    a0[0] = xf * (hi ? yf : xf);        // lo: x^2   hi: x*y
    a0[1] = hi ? xf : (yf * yf);        // lo: y^2   hi: x
    a1[0] = hi ? 0.0f : yf;             // lo: y     hi: 0 (K6)
    a1[1] = hi ? 0.0f : 1.0f;           // lo: 1     hi: 0 (K7)

    // Branchless B-fragment load offsets: low lanes read {g0,g1},{g4,g5};
    // high lanes read {g2,g3},{g6,g7} — g6/g7 are the stored zero padding.
    const int boff0 = hi ? 2 : 0;
    const int boff1 = hi ? 6 : 4;

    const size_t rowbase = (((size_t)b * TILE_H + y) * TILE_W) * N_GAUSS;

    #pragma unroll
    for (int t = 0; t < 4; ++t) {
        const int nbase = (wave * 4 + t) << 4;

        const float* gp = gpose + (size_t)((nbase + lane16) << 3);
        const v2f b0 = *(const v2f*)(gp + boff0);
        const v2f b1 = *(const v2f*)(gp + boff1);

        v8f c = {};
        c = __builtin_amdgcn_wmma_f32_16x16x4_f32(
                false, a0, false, b0, (short)0, c, false, false);
        c = __builtin_amdgcn_wmma_f32_16x16x4_f32(
                false, a1, false, b1, (short)0, c, false, false);

        const float op = opac[nbase + lane16];
        const size_t obase = rowbase + (size_t)nbase + lane16;

        #pragma unroll
        for (int r = 0; r < 8; ++r) {
            const int m = r + (hi ? 8 : 0);              // pixel row within tile
            float alpha = op * __builtin_amdgcn_exp2f(c[r]);  // bare v_exp_f32
            alpha = fminf(alpha, 1.0f);
            out[obase + (size_t)(x0 + m) * N_GAUSS] = alpha;
        }
    }
}

extern "C" void kernel_launch(void* const* d_in, const int* in_sizes, int n_in,
                              void* d_out, int out_size, void* d_ws, size_t ws_size,
                              hipStream_t stream) {
    const float* pose   = (const float*)d_in[0];
    const float* means  = (const float*)d_in[1];
    const float* quats  = (const float*)d_in[2];
    const float* scales = (const float*)d_in[3];
    const float* opac   = (const float*)d_in[4];
    float* out = (float*)d_out;
    float* G   = (float*)d_ws;   // needs B*N*8*4 = 64 KB

    gsplat_coeff_kernel<<<(B_POSE * N_GAUSS + 255) / 256, 256, 0, stream>>>(
        pose, means, quats, scales, G);

    gsplat_wmma_kernel<<<B_POSE * (TILE_H * TILE_W / 16), 256, 0, stream>>>(
        G, opac, out);
}